// softmax_40991167873103
// MI455X (gfx1250) — compile-verified
//
#include <hip/hip_runtime.h>

// Clang ext-vector types (HIP's float4 is a class; these lower to clean b128 ops
// and are accepted by the WMMA builtin / nontemporal-store builtins).
typedef float v2f __attribute__((ext_vector_type(2)));
typedef float v4f __attribute__((ext_vector_type(4)));
typedef float v8f __attribute__((ext_vector_type(8)));

#define BLOCK   256      // 8 wave32s per block
#define GRID_A  1024     // pass-1 blocks -> 1024 fp32 partials in workspace
#define GRID_C  4096     // pass-3 blocks (more blocks to hide L2 latency)
#define UNROLL  4        // 4 x b128 loads in flight per wave per iteration

// ---------------------------------------------------------------------------
// Full 32-lane sum reduction via V_WMMA_F32_16X16X4_F32.
// A-matrix 16x4 layout (2 VGPRs): VGPR0 lanes0-15 = A[m][0], lanes16-31 = A[m][2].
// acc in VGPR0, zero in VGPR1, B = all ones  =>  D[m][n] = acc[m] + acc[m+16].
// D layout: lane n holds column n; VGPR r holds row r (lanes 0-15) / r+8 (16-31).
// Sum the 8 D registers per lane + one xor-16 shuffle => full 32-lane sum.
// Requires EXEC all-ones: callers are in uniform control flow with full blocks.
// ---------------------------------------------------------------------------
__device__ __forceinline__ float wave_sum_wmma(float acc) {
    v2f a; a.x = acc;  a.y = 0.0f;     // K=1,3 contribute zero
    v2f b; b.x = 1.0f; b.y = 1.0f;     // B = ones
    v8f c = {};
    c = __builtin_amdgcn_wmma_f32_16x16x4_f32(
            /*neg_a=*/false, a, /*neg_b=*/false, b,
            /*c_mod=*/(short)0, c, /*reuse_a=*/false, /*reuse_b=*/false);
    float s = ((c[0] + c[1]) + (c[2] + c[3])) + ((c[4] + c[5]) + (c[6] + c[7]));
    s += __shfl_xor(s, 16, 32);        // join the two row-halves
    return s;                          // all 32 lanes hold the full sum
}

// Block-level sum: WMMA wave reduce, then 8 wave partials combined by thread 0.
__device__ __forceinline__ float block_sum(float acc, float* lds8) {
    float w = wave_sum_wmma(acc);
    const int lane = threadIdx.x & 31;
    const int wave = threadIdx.x >> 5;
    if (lane == 0) lds8[wave] = w;
    __syncthreads();
    float s = 0.0f;
    if (threadIdx.x == 0) {
        #pragma unroll
        for (int i = 0; i < BLOCK / 32; ++i) s += lds8[i];
    }
    return s;   // valid in thread 0 only
}

// ---------------------------------------------------------------------------
// Pass 1: per-block partial sums of exp(x). Uniform scalar trip count + x4
// unroll so each wave keeps 4 b128 loads in flight (one clause, one wait).
// Default RT cache hints leave x resident in the 192MB L2 for pass 3.
// ---------------------------------------------------------------------------
__global__ void softmax_partial_kernel(const v4f* __restrict__ x4,
                                       float* __restrict__ partials,
                                       int n4, int n) {
    __shared__ float lds8[BLOCK / 32];
    const int tid    = blockIdx.x * blockDim.x + threadIdx.x;
    const int stride = gridDim.x * blockDim.x;
    const int chunk  = stride * UNROLL;
    const int kfull  = n4 / chunk;          // uniform across the grid

    float a0 = 0.0f, a1 = 0.0f, a2 = 0.0f, a3 = 0.0f;
    int i = tid;
    for (int k = 0; k < kfull; ++k) {
        __builtin_prefetch(&x4[i + chunk], 0, 0);   // global_prefetch_b8, next chunk
        v4f v0 = x4[i];
        v4f v1 = x4[i + stride];
        v4f v2 = x4[i + 2 * stride];
        v4f v3 = x4[i + 3 * stride];
        a0 += __expf(v0.x); a1 += __expf(v0.y); a2 += __expf(v0.z); a3 += __expf(v0.w);
        a0 += __expf(v1.x); a1 += __expf(v1.y); a2 += __expf(v1.z); a3 += __expf(v1.w);
        a0 += __expf(v2.x); a1 += __expf(v2.y); a2 += __expf(v2.z); a3 += __expf(v2.w);
        a0 += __expf(v3.x); a1 += __expf(v3.y); a2 += __expf(v3.z); a3 += __expf(v3.w);
        i += chunk;
    }
    // residual float4 elements (no-op for N = 2^25 with this launch geometry)
    for (; i < n4; i += stride) {
        v4f v = x4[i];
        a0 += __expf(v.x); a1 += __expf(v.y); a2 += __expf(v.z); a3 += __expf(v.w);
    }
    // scalar tail (no-op for N % 4 == 0)
    const float* xs = (const float*)x4;
    for (int j = n4 * 4 + tid; j < n; j += stride) a0 += __expf(xs[j]);

    float acc = (a0 + a1) + (a2 + a3);
    float s = block_sum(acc, lds8);
    if (threadIdx.x == 0) partials[blockIdx.x] = s;
}

// ---------------------------------------------------------------------------
// Pass 2: single block reduces GRID_A partials -> writes 1/sum.
// ---------------------------------------------------------------------------
__global__ void softmax_finalize_kernel(const float* __restrict__ partials,
                                        float* __restrict__ inv_out,
                                        int nparts) {
    __shared__ float lds8[BLOCK / 32];
    float acc = 0.0f;
    for (int i = threadIdx.x; i < nparts; i += blockDim.x) acc += partials[i];
    float s = block_sum(acc, lds8);
    if (threadIdx.x == 0) inv_out[0] = 1.0f / s;
}

// ---------------------------------------------------------------------------
// Pass 3: out = exp(x) * inv. x re-read hits L2 (RT-resident from pass 1);
// out written with non-temporal b128 stores so 128MB of output doesn't evict x.
// ---------------------------------------------------------------------------
__global__ void softmax_normalize_kernel(const v4f* __restrict__ x4,
                                         v4f* __restrict__ out4,
                                         const float* __restrict__ inv_p,
                                         int n4, int n) {
    const float inv = inv_p[0];   // uniform scalar load
    const int tid    = blockIdx.x * blockDim.x + threadIdx.x;
    const int stride = gridDim.x * blockDim.x;
    const int chunk  = stride * UNROLL;
    const int kfull  = n4 / chunk;          // uniform across the grid

    int i = tid;
    for (int k = 0; k < kfull; ++k) {
        __builtin_prefetch(&x4[i + chunk], 0, 0);
        v4f v0 = x4[i];
        v4f v1 = x4[i + stride];
        v4f v2 = x4[i + 2 * stride];
        v4f v3 = x4[i + 3 * stride];
        v4f r0, r1, r2, r3;
        r0.x = __expf(v0.x) * inv; r0.y = __expf(v0.y) * inv;
        r0.z = __expf(v0.z) * inv; r0.w = __expf(v0.w) * inv;
        r1.x = __expf(v1.x) * inv; r1.y = __expf(v1.y) * inv;
        r1.z = __expf(v1.z) * inv; r1.w = __expf(v1.w) * inv;
        r2.x = __expf(v2.x) * inv; r2.y = __expf(v2.y) * inv;
        r2.z = __expf(v2.z) * inv; r2.w = __expf(v2.w) * inv;
        r3.x = __expf(v3.x) * inv; r3.y = __expf(v3.y) * inv;
        r3.z = __expf(v3.z) * inv; r3.w = __expf(v3.w) * inv;
        __builtin_nontemporal_store(r0, &out4[i]);
        __builtin_nontemporal_store(r1, &out4[i + stride]);
        __builtin_nontemporal_store(r2, &out4[i + 2 * stride]);
        __builtin_nontemporal_store(r3, &out4[i + 3 * stride]);
        i += chunk;
    }
    for (; i < n4; i += stride) {
        v4f v = x4[i];
        v4f r;
        r.x = __expf(v.x) * inv; r.y = __expf(v.y) * inv;
        r.z = __expf(v.z) * inv; r.w = __expf(v.w) * inv;
        __builtin_nontemporal_store(r, &out4[i]);
    }
    const float* xs = (const float*)x4;
    float*       os = (float*)out4;
    for (int j = n4 * 4 + tid; j < n; j += stride)
        __builtin_nontemporal_store(__expf(xs[j]) * inv, &os[j]);
}

// ---------------------------------------------------------------------------
extern "C" void kernel_launch(void* const* d_in, const int* in_sizes, int n_in,
                              void* d_out, int out_size, void* d_ws, size_t ws_size,
                              hipStream_t stream) {
    const float* x   = (const float*)d_in[0];
    float*       out = (float*)d_out;
    const int n  = in_sizes[0];
    const int n4 = n >> 2;

    float* partials = (float*)d_ws;          // GRID_A floats
    float* inv_sum  = partials + GRID_A;     // 1 float

    softmax_partial_kernel<<<GRID_A, BLOCK, 0, stream>>>(
        (const v4f*)x, partials, n4, n);
    softmax_finalize_kernel<<<1, BLOCK, 0, stream>>>(
        partials, inv_sum, GRID_A);
    softmax_normalize_kernel<<<GRID_C, BLOCK, 0, stream>>>(
        (const v4f*)x, (v4f*)out, inv_sum, n4, n);
}